// WaveletTransform_56075093016893
// MI455X (gfx1250) — compile-verified
//
#include <hip/hip_runtime.h>
#include <stdint.h>

// ---------------- tile geometry ----------------
#define TW    64            // subband cols per tile
#define TH     8            // subband rows per tile
#define SH   (TH + 2)       // 10  (halo)
#define SW   (TW + 2)       // 66  (halo)
#define SW_P (SW + 1)       // 67  (bank pad)
#define RAW_H (2 * SH)      // 20 input rows staged
#define RAW_W (2 * SW)      // 132 input cols staged
#define RAW_N (RAW_H * RAW_W)

typedef __attribute__((address_space(1))) int* gas1_i32;
typedef __attribute__((address_space(3))) int* las3_i32;

// CDNA5 async global -> LDS copy (per-lane LDS dest, per-lane global src).
// Tracked by ASYNCcnt.
__device__ __forceinline__ void async_ld_b32(const void* g, void* l) {
#if __has_builtin(__builtin_amdgcn_global_load_async_to_lds_b32)
  __builtin_amdgcn_global_load_async_to_lds_b32((gas1_i32)g, (las3_i32)l, 0, 0);
#else
  unsigned lds_addr = (unsigned)(uintptr_t)(las3_i32)l;
  uint64_t ga = (uint64_t)(uintptr_t)g;
  asm volatile("global_load_async_to_lds_b32 %0, %1, off"
               :: "v"(lds_addr), "v"(ga)
               : "memory");
#endif
}

__device__ __forceinline__ void wait_async0() {
#if __has_builtin(__builtin_amdgcn_s_wait_asynccnt)
  __builtin_amdgcn_s_wait_asynccnt(0);
#else
  asm volatile("s_wait_asynccnt 0" ::: "memory");
#endif
}

__device__ __forceinline__ int clamp127(int v) {
  return v < 0 ? 0 : (v > 127 ? 127 : v);
}

__global__ void __launch_bounds__(256)
haar_up_kernel(const float* __restrict__ x, float* __restrict__ out, int planes) {
  __shared__ float raw[RAW_H][RAW_W];       // staged (clamped) input tile
  __shared__ float sub[4][SH][SW_P];        // LL, LH, HL, HH with halo

  const int tid  = threadIdx.x;
  const int tile = blockIdx.x;              // 0..31
  const int p    = blockIdx.y;              // plane (b*C + c)
  const int j0   = (tile & 1) * TW;         // subband col origin (0 or 64)
  const int i0   = (tile >> 1) * TH;        // subband row origin (0..120)

  const float* xp = x + (size_t)p * 65536;  // 256*256 plane

  // ---- Phase 1: async-stage input tile into LDS.
  // Halo clamping done at the SUBBAND index level (so halo value == edge
  // subband value, matching jax.image.resize edge renormalization).
  for (int idx = tid; idx < RAW_N; idx += 256) {
    int rr = idx / RAW_W;
    int cc = idx - rr * RAW_W;
    int gs = clamp127(i0 - 1 + (rr >> 1));   // clamped subband row
    int gc = clamp127(j0 - 1 + (cc >> 1));   // clamped subband col
    int grow = 2 * gs + (rr & 1);
    int gcol = 2 * gc + (cc & 1);
    async_ld_b32(xp + grow * 256 + gcol, &raw[rr][cc]);
  }
  wait_async0();
  __syncthreads();

  // ---- Phase 2: Haar subbands (with halo) into LDS.
  for (int idx = tid; idx < SH * SW; idx += 256) {
    int si = idx / SW;
    int sj = idx - si * SW;
    float a = raw[2 * si    ][2 * sj    ];
    float b = raw[2 * si    ][2 * sj + 1];
    float c = raw[2 * si + 1][2 * sj    ];
    float d = raw[2 * si + 1][2 * sj + 1];
    sub[0][si][sj] = (a + b + c + d) * 0.5f;   // LL
    sub[1][si][sj] = (a + b - c - d) * 0.5f;   // LH
    sub[2][si][sj] = (a - b + c - d) * 0.5f;   // HL
    sub[3][si][sj] = (a - b - c + d) * 0.5f;   // HH
  }
  __syncthreads();

  // ---- Phase 3: 2x bilinear upsample (half-pixel), float4 stores.
  // Thread handles subband points (i, jj) and (i, jj+1):
  //   output rows 2I, 2I+1; output cols 2J .. 2J+3 (16B-aligned float4).
  const int i  = tid >> 5;             // 0..7
  const int jj = (tid & 31) * 2;       // 0..62, step 2
  const int li = i + 1;                // LDS row (halo offset)
  const int lj = jj + 1;               // LDS col (halo offset)

  const int gy0 = 2 * (i0 + i);
  const int gx  = 2 * (j0 + jj);
  const size_t plane_off = (size_t)p * 65536;
  const size_t TOTQ = (size_t)planes * 65536;

#pragma unroll
  for (int q = 0; q < 4; ++q) {
    const float (*S)[SW_P] = sub[q];
    float m0 = S[li - 1][lj - 1], m1 = S[li - 1][lj], m2 = S[li - 1][lj + 1], m3 = S[li - 1][lj + 2];
    float c0 = S[li    ][lj - 1], c1 = S[li    ][lj], c2 = S[li    ][lj + 1], c3 = S[li    ][lj + 2];
    float p0 = S[li + 1][lj - 1], p1 = S[li + 1][lj], p2 = S[li + 1][lj + 1], p3 = S[li + 1][lj + 2];

    // vertical blends: row 2I uses (0.25, 0.75) on rows (I-1, I);
    //                  row 2I+1 uses (0.75, 0.25) on rows (I, I+1)
    float r0a = 0.25f * m0 + 0.75f * c0;
    float r0b = 0.25f * m1 + 0.75f * c1;
    float r0c = 0.25f * m2 + 0.75f * c2;
    float r0d = 0.25f * m3 + 0.75f * c3;
    float r1a = 0.75f * c0 + 0.25f * p0;
    float r1b = 0.75f * c1 + 0.25f * p1;
    float r1c = 0.75f * c2 + 0.25f * p2;
    float r1d = 0.75f * c3 + 0.25f * p3;

    float4 o0, o1;
    o0.x = 0.25f * r0a + 0.75f * r0b;   // col 2J
    o0.y = 0.75f * r0b + 0.25f * r0c;   // col 2J+1
    o0.z = 0.25f * r0b + 0.75f * r0c;   // col 2J+2
    o0.w = 0.75f * r0c + 0.25f * r0d;   // col 2J+3
    o1.x = 0.25f * r1a + 0.75f * r1b;
    o1.y = 0.75f * r1b + 0.25f * r1c;
    o1.z = 0.25f * r1b + 0.75f * r1c;
    o1.w = 0.75f * r1c + 0.25f * r1d;

    float* ob = out + (size_t)q * TOTQ + plane_off;
    *(float4*)(ob + (size_t)gy0 * 256 + gx)       = o0;
    *(float4*)(ob + (size_t)(gy0 + 1) * 256 + gx) = o1;
  }
}

extern "C" void kernel_launch(void* const* d_in, const int* in_sizes, int n_in,
                              void* d_out, int out_size, void* d_ws, size_t ws_size,
                              hipStream_t stream) {
  (void)n_in; (void)out_size; (void)d_ws; (void)ws_size;
  const float* x = (const float*)d_in[0];
  float* out = (float*)d_out;
  const int planes = in_sizes[0] / (256 * 256);  // B*C = 512
  dim3 grid(32, planes);                         // 32 tiles per 256x256 plane
  haar_up_kernel<<<grid, 256, 0, stream>>>(x, out, planes);
}